// PsiU_10986526343471
// MI455X (gfx1250) — compile-verified
//
#include <hip/hip_runtime.h>
#include <hip/hip_bf16.h>
#include <math.h>

// Problem constants
#define NN   1024   // N
#define MM   1024   // M
#define NXI  2048
#define LL   2048
#define SX   6144   // 2*NXI + L
#define EPSC 0.001f

typedef float v2f __attribute__((ext_vector_type(2)));
typedef float v8f __attribute__((ext_vector_type(8)));
typedef unsigned int u32x4 __attribute__((ext_vector_type(4)));
typedef int i32x4 __attribute__((ext_vector_type(4)));
typedef int i32x8 __attribute__((ext_vector_type(8)));

#if defined(__has_include)
#  if __has_include(<hip/amd_detail/amd_gfx1250_TDM.h>)
#    define TDM_6ARG 1
#  endif
#endif

// GEMM tiling
#define TS   128    // block tile (M and N)
#define KC   32     // K chunk depth staged in LDS
#define LDSW (TS + 4)   // LDS row stride in floats (TDM pads 4 dwords per row)

// ---------------------------------------------------------------------------
// TDM: async-load one 32-row x 128-col f32 tile (row stride SX elements) from
// global into LDS at byte offset lds_off, padding 4 dwords after every 128
// (matching an LDS row stride of 132 floats).  Issued once per calling wave;
// completion tracked with TENSORcnt.
// ---------------------------------------------------------------------------
__device__ __forceinline__ void tdm_load_tile(const float* gsrc, unsigned lds_off)
{
    const unsigned long long ga = (unsigned long long)(uintptr_t)gsrc;
    u32x4 g0;
    g0.x = 1u;                                            // count=1 (valid user D#)
    g0.y = lds_off;                                       // LDS byte address
    g0.z = (unsigned)(ga & 0xFFFFFFFFu);                  // global_addr[31:0]
    g0.w = (unsigned)((ga >> 32) & 0x1FFFFFFu)            // global_addr[56:32]
         | (2u << 30);                                    // type = 2 ("image")
    i32x8 g1;
    g1[0] = (int)((2u << 16)      // data_size = 4 bytes
                | (1u << 20)      // pad_enable
                | (6u << 22)      // pad_interval: 128 dwords
                | (3u << 25));    // pad_amount:  4 dwords
    g1[1] = 0;                                            // tensor_dim0 low16 = 0
    g1[2] = (int)0x4000;                                  // tensor_dim0 = 0x40000000
    g1[3] = (int)(0x4000u | ((unsigned)TS << 16));        // tensor_dim1 big, tile_dim0 = 128
    g1[4] = KC;                                           // tile_dim1 = 32, tile_dim2 = 0
    g1[5] = SX;                                           // tensor_dim0_stride = 6144
    g1[6] = 0;
    g1[7] = 0;
    i32x4 gz = {};
#ifdef TDM_6ARG
    i32x8 gz8 = {};
    __builtin_amdgcn_tensor_load_to_lds(g0, g1, gz, gz, gz8, 0);
#else
    __builtin_amdgcn_tensor_load_to_lds(g0, g1, gz, gz, 0);
#endif
}

// ---------------------------------------------------------------------------
// WMMA f32 GEMM:  C[n x n] = sum over slabs of X[:,ca:ca+n]^T @ X[:,cb:cb+n]
// 128x128 output tile per 256-thread (8-wave) block; each wave owns a 32x64
// sub-tile (2 A-frags + 4 B-frags amortized over 8 WMMAs -> 1.5 LDS dwords
// per lane per WMMA).  K staged through LDS in 32-deep chunks by the Tensor
// Data Mover with double buffering; wave 0 pipelines one chunk ahead.
// lowerOnly=1 computes only tiles with tileRow >= tileCol (symmetric result;
// upper mirrored in an epilogue).
// ---------------------------------------------------------------------------
__global__ void __launch_bounds__(256)
wmma_xtx_kernel(const float* __restrict__ X, float* __restrict__ C,
                int lda, int K, int n,
                int colA0, int colB0, int colA1, int colB1, int nslab,
                int lowerOnly)
{
    const int tRow = blockIdx.y;
    const int tCol = blockIdx.x;
    if (lowerOnly && tRow < tCol) return;

    __shared__ float As[2][KC][LDSW];
    __shared__ float Bs[2][KC][LDSW];

    const int tid  = threadIdx.x;
    const int wave = tid >> 5;
    const int lane = tid & 31;
    const int mBase = (wave & 3) * 32;    // 4 row-strips of 32
    const int nBase = (wave >> 2) * 64;   // 2 col-halves of 64

    v8f acc[2][4] = {};

    const int m0 = tRow * TS;
    const int n0 = tCol * TS;

    const int kq = (lane >> 4) * 2;       // lanes 16-31 hold K+2/K+3 (ISA 7.12.2)
    const int ml = lane & 15;

    const int nkc = K / KC;               // chunks per slab
    const int nch = nslab * nkc;          // total chunks

    // Prologue: chunk 0 into buffer 0.
    if (wave == 0) {
        tdm_load_tile(X + colA0 + m0, (unsigned)(uintptr_t)&As[0][0][0]);
        tdm_load_tile(X + colB0 + n0, (unsigned)(uintptr_t)&Bs[0][0][0]);
    }

    for (int c = 0; c < nch; ++c) {
        const int cur = c & 1;
        if (wave == 0) {
            if (c + 1 < nch) {
                const int ns  = (c + 1) / nkc;
                const int nkb = ((c + 1) % nkc) * KC;
                const int ca  = ns ? colA1 : colA0;
                const int cb  = ns ? colB1 : colB0;
                tdm_load_tile(X + (size_t)nkb * lda + ca + m0,
                              (unsigned)(uintptr_t)&As[cur ^ 1][0][0]);
                tdm_load_tile(X + (size_t)nkb * lda + cb + n0,
                              (unsigned)(uintptr_t)&Bs[cur ^ 1][0][0]);
                __builtin_amdgcn_s_wait_tensorcnt((short)2);  // current chunk landed
            } else {
                __builtin_amdgcn_s_wait_tensorcnt((short)0);
            }
        }
        __syncthreads();                  // publish TDM data to all waves

        #pragma unroll
        for (int kk = 0; kk < KC; kk += 4) {
            v2f a[2];
            #pragma unroll
            for (int mt = 0; mt < 2; ++mt) {
                a[mt].x = As[cur][kk + kq][mBase + mt * 16 + ml];
                a[mt].y = As[cur][kk + kq + 1][mBase + mt * 16 + ml];
            }
            #pragma unroll
            for (int nt = 0; nt < 4; ++nt) {
                v2f b;
                b.x = Bs[cur][kk + kq][nBase + nt * 16 + ml];
                b.y = Bs[cur][kk + kq + 1][nBase + nt * 16 + ml];
                #pragma unroll
                for (int mt = 0; mt < 2; ++mt)
                    acc[mt][nt] = __builtin_amdgcn_wmma_f32_16x16x4_f32(
                        false, a[mt], false, b, (short)0, acc[mt][nt], false, false);
            }
        }
        __syncthreads();                  // all waves done before buffer reuse
    }

    // C/D layout: VGPR r -> M = r (lanes 0-15), M = r+8 (lanes 16-31); N = lane&15.
    const int rowHalf = (lane >> 4) * 8;
    const int nl = lane & 15;
    #pragma unroll
    for (int mt = 0; mt < 2; ++mt) {
        #pragma unroll
        for (int nt = 0; nt < 4; ++nt) {
            #pragma unroll
            for (int r = 0; r < 8; ++r) {
                const int gr = m0 + mBase + mt * 16 + r + rowHalf;
                const int gc = n0 + nBase + nt * 16 + nl;
                C[(size_t)gr * n + gc] = acc[mt][nt][r];
            }
        }
    }
}

// ---------------------------------------------------------------------------
// E = 0.5*S + eps*I + 0.5*(Y - Y^T), mirroring S's lower triangle (in place).
// ---------------------------------------------------------------------------
__global__ void make_E_kernel(float* S, const float* __restrict__ Y, int n)
{
    const int j = blockIdx.x * blockDim.x + threadIdx.x;
    const int i = blockIdx.y;
    if (i >= n || j >= n || j > i) return;
    const float s    = S[(size_t)i * n + j];
    const float asym = 0.5f * (Y[(size_t)i * n + j] - Y[(size_t)j * n + i]);
    const float d    = (i == j) ? EPSC : 0.0f;
    S[(size_t)i * n + j] = 0.5f * s + asym + d;
    if (i != j) S[(size_t)j * n + i] = 0.5f * s - asym;
}

// Lam[i] = 0.5*(H22[i][i] + eps)
__global__ void lam_kernel(const float* __restrict__ H, float* lam, int n)
{
    const int i = blockIdx.x * blockDim.x + threadIdx.x;
    if (i < n) lam[i] = 0.5f * (H[(size_t)i * n + i] + EPSC);
}

// ---------------------------------------------------------------------------
// Row-dot GEMV: out[r] = (addv?addv[r]:0) + dot(A[r, coff:coff+cols], x).
// One wave32 per output row; coalesced row reads; shuffle reduce.
// ---------------------------------------------------------------------------
__global__ void rowdot_kernel(const float* __restrict__ A, int lda, int coff,
                              int cols, const float* __restrict__ x,
                              const float* __restrict__ addv,
                              float* __restrict__ out, int rows)
{
    const int gid  = blockIdx.x * blockDim.x + threadIdx.x;
    const int r    = gid >> 5;
    const int lane = threadIdx.x & 31;
    if (r >= rows) return;
    const float* row = A + (size_t)r * lda + coff;
    float s = 0.0f;
    for (int c = lane; c < cols; c += 32) s += row[c] * x[c];
    for (int off = 16; off > 0; off >>= 1) s += __shfl_down(s, off, 32);
    if (lane == 0) out[r] = (addv ? addv[r] : 0.0f) + s;
}

// ---------------------------------------------------------------------------
// Column GEMV: out[c] = sgn * sum_k X[k, coff+c]*v[k] + dot(R[c,:rcols], w).
// One thread per output column; X reads coalesced across threads.
// ---------------------------------------------------------------------------
__global__ void colgemv_kernel(const float* __restrict__ X, int lda, int coff,
                               int K, const float* __restrict__ v, float sgn,
                               const float* __restrict__ R, int rcols,
                               const float* __restrict__ w,
                               float* __restrict__ out, int n)
{
    const int c = blockIdx.x * blockDim.x + threadIdx.x;
    if (c >= n) return;
    const float* p = X + coff + c;
    float s = 0.0f;
    for (int k = 0; k < K; ++k) s += p[(size_t)k * lda] * v[k];
    float s2 = 0.0f;
    const float* rr = R + (size_t)c * rcols;
    for (int j = 0; j < rcols; ++j) s2 += rr[j] * w[j];
    out[c] = sgn * s + s2;
}

// ---------------------------------------------------------------------------
// Sequential scan: eps[i] = tanh((base[i] - sum_{j<i} H22[i][j]*eps[j]) / lam[i])
// Single 1024-thread workgroup; eps cached in LDS; tree reduction per row.
// ---------------------------------------------------------------------------
__global__ void __launch_bounds__(1024)
scan_kernel(const float* __restrict__ H22, const float* __restrict__ base,
            const float* __restrict__ lam, float* __restrict__ eps_out, int n)
{
    __shared__ float red[1024];
    __shared__ float epsS[LL];
    const int tid = threadIdx.x;
    for (int j = tid; j < n; j += 1024) epsS[j] = 0.0f;
    __syncthreads();
    for (int i = 0; i < n; ++i) {
        const float* row = H22 + (size_t)i * n;
        float p = 0.0f;
        for (int j = tid; j < i; j += 1024) p += row[j] * epsS[j];
        red[tid] = p;
        __syncthreads();
        for (int s = 512; s > 0; s >>= 1) {
            if (tid < s) red[tid] += red[tid + s];
            __syncthreads();
        }
        if (tid == 0) epsS[i] = tanhf((base[i] - red[0]) / lam[i]);
        __syncthreads();
    }
    for (int j = tid; j < n; j += 1024) eps_out[j] = epsS[j];
}

// ---------------------------------------------------------------------------
// In-place LU (no pivoting; E's symmetric part is PD). Single workgroup.
// ---------------------------------------------------------------------------
__global__ void __launch_bounds__(1024)
lu_kernel(float* __restrict__ E, int n)
{
    const int tid  = threadIdx.x;
    const int wid  = tid >> 5;
    const int lane = tid & 31;
    for (int k = 0; k < n - 1; ++k) {
        const float inv = 1.0f / E[(size_t)k * n + k];
        for (int i = k + 1 + tid; i < n; i += 1024) E[(size_t)i * n + k] *= inv;
        __syncthreads();
        for (int i = k + 1 + wid; i < n; i += 32) {
            const float lik = E[(size_t)i * n + k];
            for (int j = k + 1 + lane; j < n; j += 32)
                E[(size_t)i * n + j] -= lik * E[(size_t)k * n + j];
        }
        __syncthreads();
    }
}

// Forward (unit-lower) then backward (upper) substitution.  Single workgroup.
__global__ void __launch_bounds__(1024)
trisolve_kernel(const float* __restrict__ E, const float* __restrict__ b,
                float* __restrict__ x_out, int n)
{
    __shared__ float red[1024];
    __shared__ float v[NXI];
    const int tid = threadIdx.x;
    for (int i = 0; i < n; ++i) {                 // L y = b
        const float* row = E + (size_t)i * n;
        float p = 0.0f;
        for (int j = tid; j < i; j += 1024) p += row[j] * v[j];
        red[tid] = p;
        __syncthreads();
        for (int s = 512; s > 0; s >>= 1) {
            if (tid < s) red[tid] += red[tid + s];
            __syncthreads();
        }
        if (tid == 0) v[i] = b[i] - red[0];
        __syncthreads();
    }
    for (int i = n - 1; i >= 0; --i) {            // U x = y
        const float* row = E + (size_t)i * n;
        float p = 0.0f;
        for (int j = i + 1 + tid; j < n; j += 1024) p += row[j] * v[j];
        red[tid] = p;
        __syncthreads();
        for (int s = 512; s > 0; s >>= 1) {
            if (tid < s) red[tid] += red[tid + s];
            __syncthreads();
        }
        if (tid == 0) v[i] = (v[i] - red[0]) / row[i];
        __syncthreads();
    }
    for (int j = tid; j < n; j += 1024) x_out[j] = v[j];
}

// u = 8 + 4.8*sigmoid(C2@xi + D21@eps + D22@w + bu).  One wave per output.
__global__ void u_kernel(const float* __restrict__ C2, const float* __restrict__ D21,
                         const float* __restrict__ D22, const float* __restrict__ xi,
                         const float* __restrict__ eps, const float* __restrict__ w,
                         const float* __restrict__ bu, float* __restrict__ out,
                         int m, int nxi, int nn)
{
    const int gid  = blockIdx.x * blockDim.x + threadIdx.x;
    const int r    = gid >> 5;
    const int lane = threadIdx.x & 31;
    if (r >= m) return;
    float s = 0.0f;
    const float* c2 = C2 + (size_t)r * nxi;
    const float* d1 = D21 + (size_t)r * nxi;
    for (int c = lane; c < nxi; c += 32) s += c2[c] * xi[c] + d1[c] * eps[c];
    const float* d2 = D22 + (size_t)r * nn;
    for (int c = lane; c < nn; c += 32) s += d2[c] * w[c];
    for (int off = 16; off > 0; off >>= 1) s += __shfl_down(s, off, 32);
    if (lane == 0) {
        const float vv = s + bu[r];
        out[r] = 8.0f + 4.8f / (1.0f + expf(-vv));
    }
}

// ---------------------------------------------------------------------------
extern "C" void kernel_launch(void* const* d_in, const int* in_sizes, int n_in,
                              void* d_out, int out_size, void* d_ws, size_t ws_size,
                              hipStream_t stream)
{
    (void)in_sizes; (void)n_in; (void)out_size; (void)ws_size;
    const float* w   = (const float*)d_in[1];
    const float* xi  = (const float*)d_in[2];
    const float* X   = (const float*)d_in[3];
    const float* Y   = (const float*)d_in[4];
    const float* B2  = (const float*)d_in[5];
    const float* C2  = (const float*)d_in[6];
    const float* D21 = (const float*)d_in[7];
    const float* D22 = (const float*)d_in[8];
    const float* D12 = (const float*)d_in[9];
    const float* bu  = (const float*)d_in[10];
    float* out = (float*)d_out;

    float* Mbuf = (float*)d_ws;                        // 2048x2048: H22, then S/E
    float* y    = Mbuf + (size_t)NXI * NXI;            // 6144
    float* z    = y + SX;                              // 6144
    float* base = z + SX;                              // 2048
    float* lam  = base + NXI;                          // 2048
    float* eps  = lam + NXI;                           // 2048
    float* Exi  = eps + NXI;                           // 2048

    // 1) y = X1 @ xi          (6144 rows, wave per row)
    rowdot_kernel<<<(SX * 32) / 256, 256, 0, stream>>>(X, SX, 0, NXI, xi, nullptr, y, SX);
    // 2) base = -X2^T y + D12 @ w
    colgemv_kernel<<<NXI / 256, 256, 0, stream>>>(X, SX, NXI, SX, y, -1.0f, D12, NN, w, base, NXI);
    // 3) H22 = X2^T X2  (lower-triangular tiles only) -- fp32 WMMA + TDM
    wmma_xtx_kernel<<<dim3(NXI / TS, NXI / TS), 256, 0, stream>>>(
        X, Mbuf, SX, SX, NXI, NXI, NXI, 0, 0, 1, 1);
    // 4) Lam
    lam_kernel<<<NXI / 256, 256, 0, stream>>>(Mbuf, lam, NXI);
    // 5) sequential tanh scan
    scan_kernel<<<1, 1024, 0, stream>>>(Mbuf, base, lam, eps, NXI);
    // 6) z = y + X2 @ eps
    rowdot_kernel<<<(SX * 32) / 256, 256, 0, stream>>>(X, SX, NXI, NXI, eps, y, z, SX);
    // 7) Exi = X3^T z + B2 @ w
    colgemv_kernel<<<NXI / 256, 256, 0, stream>>>(X, SX, 2 * NXI, SX, z, 1.0f, B2, NN, w, Exi, NXI);
    // 8) S = X1^T X1 + X3^T X3 (lower tiles, two K-slabs) -- fp32 WMMA + TDM
    wmma_xtx_kernel<<<dim3(NXI / TS, NXI / TS), 256, 0, stream>>>(
        X, Mbuf, SX, SX, NXI, 0, 0, 2 * NXI, 2 * NXI, 2, 1);
    // 9) E = 0.5*S + eps*I + 0.5*(Y - Y^T), mirrored in place
    make_E_kernel<<<dim3(NXI / 256, NXI), 256, 0, stream>>>(Mbuf, Y, NXI);
    // 10) LU(E) in place
    lu_kernel<<<1, 1024, 0, stream>>>(Mbuf, NXI);
    // 11) xi_next = E^{-1} Exi  -> out[1024:3072]
    trisolve_kernel<<<1, 1024, 0, stream>>>(Mbuf, Exi, out + MM, NXI);
    // 12) u -> out[0:1024]
    u_kernel<<<(MM * 32) / 256, 256, 0, stream>>>(C2, D21, D22, xi, eps, w, bu, out, MM, NXI, NN);
}